// UMNNMAF_22978075033891
// MI455X (gfx1250) — compile-verified
//
#include <hip/hip_runtime.h>
#include <math.h>

// ---------------------------------------------------------------------------
// UMNN MAF forward for MI455X (gfx1250, wave32, WMMA).
//   B=4096, D=32, M1=8, H=64, S=21 quadrature nodes.
// Phase 1: h = tanh(x @ (We*mask)^T + be)  -> d_ws  (4096 x 256 fp32, 4 MB)
//          via V_WMMA_F32_16X16X4_F32; mask applied as v_cndmask on loaded
//          values (unconditional b64 loads, no EXEC predication).
// Phase 2: per (d, 16-row b tile): one-shot WMMA GEMM for the s-invariant
//          h-features (16x8x64), then 21 quadrature steps as rank-1 updates
//          + ELU + W2-dot (half-wave shuffle reduce) + outer ELU + CC weight.
// ---------------------------------------------------------------------------

typedef float v2f __attribute__((ext_vector_type(2)));
typedef float v8f __attribute__((ext_vector_type(8)));

#define PI_F 3.14159265358979323846f

__device__ __forceinline__ float elu_f(float v) {
    return v > 0.0f ? v : (__expf(v) - 1.0f);
}

// ---------------------------------------------------------------------------
// Phase 1: h[b, j] = tanh( sum_k x[b,k] * We[j,k] * (k < j%32) + be[j] )
// grid: (256 M-tiles of 16 rows, 2) ; block 256 (8 waves); wave -> one 16x16 tile
// ---------------------------------------------------------------------------
__global__ __launch_bounds__(256)
void umnn_h_kernel(const float* __restrict__ x,    // (4096, 32)
                   const float* __restrict__ We,   // (256, 32)
                   const float* __restrict__ be,   // (256,)
                   float* __restrict__ h)          // (4096, 256)
{
    const int wave  = threadIdx.x >> 5;
    const int lane  = threadIdx.x & 31;
    const int lhalf = lane >> 4;          // 0: lanes 0-15, 1: lanes 16-31
    const int l16   = lane & 15;

    const int b0 = blockIdx.x * 16;                 // 16 batch rows (M)
    const int n0 = (blockIdx.y * 8 + wave) * 16;    // 16 output cols (N)
    const int n  = n0 + l16;                        // this lane's column j
    const int jmask = n & 31;                       // j % 32 (mask threshold)

    // Per-lane row bases; each K-step reads an 8-byte-aligned float2.
    const float* xrow  = x  + (b0 + l16) * 32 + 2 * lhalf;
    const float* werow = We + n * 32 + 2 * lhalf;

    // C init = bias broadcast along M
    v8f acc;
    const float bias = be[n];
#pragma unroll
    for (int i = 0; i < 8; ++i) acc[i] = bias;

    // K = 32 -> 8 chained 16x16x4 f32 WMMAs
#pragma unroll
    for (int kk = 0; kk < 8; ++kk) {
        const int kp = kk * 4 + 2 * lhalf;   // this half-wave's K pair
        // A(16x4): lane l16 = row m, VGPR0/1 = K pair  (unconditional b64 load)
        const v2f a = *(const v2f*)(xrow + kk * 4);
        // B(4x16): lane l16 = col n; load unconditionally, mask via cndmask
        const v2f w = *(const v2f*)(werow + kk * 4);
        v2f bm;
        bm[0] = (kp     < jmask) ? w[0] : 0.0f;
        bm[1] = (kp + 1 < jmask) ? w[1] : 0.0f;
        acc = __builtin_amdgcn_wmma_f32_16x16x4_f32(
            false, a, false, bm, (short)0, acc, false, false);
    }

    // D(16x16): VGPR i -> row (i + 8*lhalf), col n ; apply tanh and store
#pragma unroll
    for (int i = 0; i < 8; ++i) {
        const int m = i + 8 * lhalf;
        h[(b0 + m) * 256 + n] = tanhf(acc[i]);
    }
}

// ---------------------------------------------------------------------------
// Phase 2: main integration kernel.
// grid: (32 d, 32 b-chunks of 128) ; block 256 ; wave -> (d, 16 b rows)
// ---------------------------------------------------------------------------
__global__ __launch_bounds__(256)
void umnn_main_kernel(const float* __restrict__ x,        // (4096, 32)
                      const float* __restrict__ x0,       // (4096, 32)
                      const float* __restrict__ h,        // (4096, 256) ws
                      const float* __restrict__ W1,       // (32, 9, 64)
                      const float* __restrict__ b1,       // (32, 64)
                      const float* __restrict__ W2,       // (32, 64)
                      const float* __restrict__ b2,       // (32,)
                      const float* __restrict__ scaling,  // (32,)
                      float* __restrict__ out)            // (4096, 32)
{
    const int wave  = threadIdx.x >> 5;
    const int lane  = threadIdx.x & 31;
    const int lhalf = lane >> 4;
    const int l16   = lane & 15;

    const int d  = blockIdx.x;                       // feature dim, fixed per block
    const int b0 = blockIdx.y * 128 + wave * 16;     // this wave's 16 batch rows

    // --- Clenshaw-Curtis nodes + weights, computed once per block into LDS ---
    __shared__ float s_cc[21];   // quadrature weights
    __shared__ float s_cs[21];   // (step+1)/2
    if (threadIdx.x < 21) {
        const int s = threadIdx.x;
        const float step = cosf((float)s * PI_F / 20.0f);
        s_cs[s] = (step + 1.0f) * 0.5f;
        float cc = 0.0f;
#pragma unroll
        for (int k = 0; k <= 20; k += 2) {
            const float Wk  = (k == 0) ? 1.0f : 2.0f / (1.0f - (float)(k * k));
            const float lam = (s == 0 || s == 20)
                                ? 0.5f
                                : cosf((float)(k * s) * PI_F / 20.0f);
            cc += lam * Wk;
        }
        s_cc[s] = 0.1f * cc;     // * 2/nb_steps
    }
    __syncthreads();

    const float* W1d = W1 + d * (9 * 64);
    const float* b1d = b1 + d * 64;
    const float* W2d = W2 + d * 64;

    // Per-N-tile accumulators (4 tiles cover H=64), init = b1 broadcast.
    v8f   acc[4];
    float w0[4], w2[4];
#pragma unroll
    for (int t = 0; t < 4; ++t) {
        const int n  = t * 16 + l16;
        const float bv = b1d[n];
#pragma unroll
        for (int i = 0; i < 8; ++i) acc[t][i] = bv;
        w0[t] = W1d[n];          // W1[d][0][n]  (X column weights)
        w2[t] = W2d[n];          // W2[d][n][0]
    }

    // s-invariant GEMM: A(16x8) = h-features, B(8x64) = W1[d][1..8][:].
    // K=8 -> 2 chained WMMAs per N-tile.
    const float* hrow = h + (b0 + l16) * 256 + d;           // A-side base
    const float* wrow = W1d + 64 + (2 * lhalf) * 64 + l16;  // B-side base (row 1+)
#pragma unroll
    for (int kk = 0; kk < 2; ++kk) {
        const int i0 = kk * 4 + 2 * lhalf;   // feature pair for this half-wave
        v2f a;
        a[0] = hrow[(i0    ) * 32];
        a[1] = hrow[(i0 + 1) * 32];
#pragma unroll
        for (int t = 0; t < 4; ++t) {
            v2f bm;
            bm[0] = wrow[kk * 4 * 64           + t * 16];
            bm[1] = wrow[kk * 4 * 64 + 64      + t * 16];
            acc[t] = __builtin_amdgcn_wmma_f32_16x16x4_f32(
                false, a, false, bm, (short)0, acc[t], false, false);
        }
    }

    // Per-row X components. Accumulator VGPR j holds row m = j + 8*lhalf.
    float dxv[8], x0v[8];
#pragma unroll
    for (int j = 0; j < 8; ++j) {
        const int bb = b0 + j + 8 * lhalf;
        const float xv  = x [bb * 32 + d];
        const float x0s = x0[bb * 32 + d];
        x0v[j] = x0s;
        dxv[j] = xv - x0s;
    }
    const float b2s = b2[d];

    float accD[8];
#pragma unroll
    for (int j = 0; j < 8; ++j) accD[j] = 0.0f;

    // --- 21 quadrature nodes: rank-1 update + ELU + W2 dot + outer ELU ---
    for (int s = 0; s < 21; ++s) {
        const float cs = s_cs[s];
        const float cc = s_cc[s];
#pragma unroll
        for (int j = 0; j < 8; ++j) {
            const float tx = fmaf(dxv[j], cs, x0v[j]);   // X[s, row(j), d]
            float pj = 0.0f;
#pragma unroll
            for (int t = 0; t < 4; ++t) {
                const float hid = fmaf(tx, w0[t], acc[t][j]);
                pj = fmaf(elu_f(hid), w2[t], pj);
            }
            // Reduce over the 16 lanes of this half-wave (n = 0..15 per tile).
            pj += __shfl_xor(pj, 1, 16);
            pj += __shfl_xor(pj, 2, 16);
            pj += __shfl_xor(pj, 4, 16);
            pj += __shfl_xor(pj, 8, 16);
            const float dz = elu_f(pj + b2s) + 1.0f;
            accD[j] = fmaf(cc, dz, accD[j]);
        }
    }

    // One lane per half-wave writes its 8 rows.
    if (l16 == 0) {
        const float se = __expf(scaling[d]);
#pragma unroll
        for (int j = 0; j < 8; ++j) {
            const int bb = b0 + j + 8 * lhalf;
            const float z0 = h[bb * 256 + d];            // h[b, 0, d]
            out[bb * 32 + d] = se * fmaf(dxv[j] * 0.5f, accD[j], z0);
        }
    }
}

// ---------------------------------------------------------------------------
extern "C" void kernel_launch(void* const* d_in, const int* in_sizes, int n_in,
                              void* d_out, int out_size, void* d_ws, size_t ws_size,
                              hipStream_t stream) {
    const float* x       = (const float*)d_in[0];
    const float* x0      = (const float*)d_in[1];
    const float* We      = (const float*)d_in[2];
    const float* be      = (const float*)d_in[3];
    const float* W1      = (const float*)d_in[4];
    const float* b1      = (const float*)d_in[5];
    const float* W2      = (const float*)d_in[6];
    const float* b2      = (const float*)d_in[7];
    const float* scaling = (const float*)d_in[8];
    // d_in[9] = nb_steps (compile-time constant 20 here)
    float* out = (float*)d_out;
    float* h   = (float*)d_ws;   // 4096*256*4 = 4 MB scratch

    dim3 g1(256, 2);   // 256 M-tiles x (2 * 8 waves) N-tiles
    umnn_h_kernel<<<g1, 256, 0, stream>>>(x, We, be, h);

    dim3 g2(32, 32);   // 32 d x 32 chunks of 128 batch rows
    umnn_main_kernel<<<g2, 256, 0, stream>>>(x, x0, h, W1, b1, W2, b2,
                                             scaling, out);
}